// Akima_55482387529902
// MI455X (gfx1250) — compile-verified
//
#include <hip/hip_runtime.h>
#include <hip/hip_bf16.h>

// Akima spline interpolation, 256 fixed nodes on [0,1], fp32 in/out.
// HBM-bound: 8 B/element * 33.55M elements = 268 MB  ->  ~11.5 us @ 23.3 TB/s.
// Strategy:
//  - per-interval cubic coefficients precomputed into an LDS float4 table
//    (1 ds_load_b128 + 3 fmaf per element in the hot loop)
//  - b128 non-temporal streaming loads/stores (stream 268 MB once, keep L2 clean)
//  - async global->LDS staging of the node table (ASYNCcnt path)
//  - grid-stride persistent blocks, wave32-native 256-thread blocks, 2x unroll

typedef float v4f __attribute__((ext_vector_type(4)));

#define AKIMA_NODES 256
#define AKIMA_NM1   255.0f          // (NODES-1)
#define AKIMA_H     (1.0f / 255.0f) // node spacing

// result = c0 + u*(c1 + u*(c2 + u*c3)) over interval idx
__device__ __forceinline__ float akima_eval(float x, const v4f* coef) {
    float xc = fminf(fmaxf(x, 0.0f), 1.0f);   // v_med3-style clamp
    float tt = xc * AKIMA_NM1;                // t in [0, 255]
    int   idx = (int)tt;                      // tt >= 0 -> trunc == floor
    idx = idx > (AKIMA_NODES - 2) ? (AKIMA_NODES - 2) : idx;
    float u = tt - (float)idx;
    v4f  c  = coef[idx];                      // ds_load_b128
    return fmaf(u, fmaf(u, fmaf(u, c.w, c.z), c.y), c.x);
}

__global__ __launch_bounds__(256) void Akima_55482387529902_kernel(
    const float* __restrict__ input,
    const float* __restrict__ value,
    float* __restrict__ out,
    long long n)
{
    __shared__ float sv[AKIMA_NODES];          // node values
    __shared__ float me[AKIMA_NODES + 3];      // extended secant slopes
    __shared__ float hs[AKIMA_NODES];          // h * akima slope per node
    __shared__ v4f   coef[AKIMA_NODES - 1];    // per-interval cubic coefficients

    const int t = threadIdx.x;

    // ---- Stage node values into LDS -------------------------------------
#if defined(__gfx1250__) && __has_builtin(__builtin_amdgcn_global_load_async_to_lds_b32) && __has_builtin(__builtin_amdgcn_s_wait_asynccnt)
    // CDNA5 async global->LDS copy path (ASYNCcnt tracked).
    __builtin_amdgcn_global_load_async_to_lds_b32(
        (__attribute__((address_space(1))) int*)(value + t),
        (__attribute__((address_space(3))) int*)(sv + t),
        /*offset=*/0, /*cpol=*/0);
    __builtin_amdgcn_s_wait_asynccnt(0);
#else
    sv[t] = value[t];
#endif
    __syncthreads();

    // ---- Interior secant slopes: m[j] = (v[j+1]-v[j]) / h, me[j+2] = m[j]
    if (t < AKIMA_NODES - 1)
        me[t + 2] = (sv[t + 1] - sv[t]) * AKIMA_NM1;
    __syncthreads();

    if (t == 0) {
        float m0  = me[2];
        float m1  = me[3];
        float mm1 = 2.0f * m0 - m1;          // m[-1]
        me[1] = mm1;
        me[0] = 2.0f * mm1 - m0;             // m[-2]
        float mL  = me[AKIMA_NODES];         // m[n-2]
        float mL1 = me[AKIMA_NODES - 1];     // m[n-3]
        float mp1 = 2.0f * mL - mL1;         // m[n-1]
        me[AKIMA_NODES + 1] = mp1;
        me[AKIMA_NODES + 2] = 2.0f * mp1 - mL;
    }
    __syncthreads();

    // ---- Akima node slopes, pre-scaled by h ------------------------------
    {
        float e0 = me[t];
        float e1 = me[t + 1];
        float e2 = me[t + 2];
        float e3 = me[t + 3];
        float w1 = fabsf(e3 - e2);
        float w2 = fabsf(e1 - e0);
        float den = w1 + w2;
        float s = (den > 0.0f) ? (w1 * e1 + w2 * e2) / den
                               : 0.5f * (e1 + e2);
        hs[t] = s * AKIMA_H;
    }
    __syncthreads();

    // ---- Per-interval cubic coefficients {c0,c1,c2,c3} -------------------
    if (t < AKIMA_NODES - 1) {
        float v0 = sv[t];
        float v1 = sv[t + 1];
        float b0 = hs[t];
        float b1 = hs[t + 1];
        float d  = v1 - v0;
        v4f c;
        c.x = v0;                          // c0
        c.y = b0;                          // c1
        c.z = 3.0f * d - 2.0f * b0 - b1;   // c2
        c.w = b0 + b1 - 2.0f * d;          // c3
        coef[t] = c;
    }
    __syncthreads();

    // ---- Streaming interpolation: float4 non-temporal loads/stores -------
    const long long n4     = n >> 2;
    const long long stride = (long long)gridDim.x * (long long)blockDim.x;
    const long long base   = (long long)blockIdx.x * (long long)blockDim.x + t;

    const v4f* __restrict__ in4  = (const v4f*)input;
    v4f*       __restrict__ out4 = (v4f*)out;

    long long i = base;
    // 2x unrolled: two independent b128 NT loads in flight per iteration
    for (; i + stride < n4; i += 2 * stride) {
        v4f xa = __builtin_nontemporal_load(in4 + i);
        v4f xb = __builtin_nontemporal_load(in4 + i + stride);
        v4f ra, rb;
        ra.x = akima_eval(xa.x, coef);
        ra.y = akima_eval(xa.y, coef);
        ra.z = akima_eval(xa.z, coef);
        ra.w = akima_eval(xa.w, coef);
        rb.x = akima_eval(xb.x, coef);
        rb.y = akima_eval(xb.y, coef);
        rb.z = akima_eval(xb.z, coef);
        rb.w = akima_eval(xb.w, coef);
        __builtin_nontemporal_store(ra, out4 + i);
        __builtin_nontemporal_store(rb, out4 + i + stride);
    }
    for (; i < n4; i += stride) {
        v4f x = __builtin_nontemporal_load(in4 + i);
        v4f r;
        r.x = akima_eval(x.x, coef);
        r.y = akima_eval(x.y, coef);
        r.z = akima_eval(x.z, coef);
        r.w = akima_eval(x.w, coef);
        __builtin_nontemporal_store(r, out4 + i);
    }

    // ---- Scalar tail (n not divisible by 4) ------------------------------
    for (long long j = (n4 << 2) + base; j < n; j += stride) {
        float r = akima_eval(input[j], coef);
        __builtin_nontemporal_store(r, out + j);
    }
}

extern "C" void kernel_launch(void* const* d_in, const int* in_sizes, int n_in,
                              void* d_out, int out_size, void* d_ws, size_t ws_size,
                              hipStream_t stream) {
    const float* input = (const float*)d_in[0];   // (32,1024,1024) fp32
    const float* value = (const float*)d_in[1];   // (256,) fp32
    float* out = (float*)d_out;

    long long n  = (long long)out_size;
    long long n4 = n >> 2;

    const int threads = 256;                       // 8 wave32 waves per block
    long long want = (n4 + threads - 1) / threads;
    if (want < 1) want = 1;
    int blocks = (int)(want > 4096 ? 4096 : want); // grid-stride persistent blocks

    Akima_55482387529902_kernel<<<blocks, threads, 0, stream>>>(input, value, out, n);
}